// pose3d_calibration_45707041964540
// MI455X (gfx1250) — compile-verified
//
#include <hip/hip_runtime.h>
#include <hip/hip_bf16.h>

#define NJ       21
#define W_SYM    1.0f
#define W_PROJ   0.5f
#define FOCAL_C  512.0f
#define PX_C     512.0f
#define PY_C     288.0f

typedef __attribute__((ext_vector_type(2))) float v2f;
typedef __attribute__((ext_vector_type(8))) float v8f;

// ds_swizzle_b32 xor-swap add (group-of-32 mode: offset = xor<<10 | 0x1f)
__device__ __forceinline__ float swz_add_1(float v) {
    return v + __int_as_float(__builtin_amdgcn_ds_swizzle(__float_as_int(v), 0x041f));
}
__device__ __forceinline__ float swz_add_2(float v) {
    return v + __int_as_float(__builtin_amdgcn_ds_swizzle(__float_as_int(v), 0x081f));
}
__device__ __forceinline__ float swz_add_4(float v) {
    return v + __int_as_float(__builtin_amdgcn_ds_swizzle(__float_as_int(v), 0x101f));
}
__device__ __forceinline__ float swz_add_8(float v) {
    return v + __int_as_float(__builtin_amdgcn_ds_swizzle(__float_as_int(v), 0x201f));
}

// Wave32 reduction: first stage via V_WMMA_F32_16X16X4_F32.
// A = ones(16x4); B: b = {partial, 0} -> B[0][j]=p_j (lanes 0-15), B[2][j]=p_{j+16}.
// D[i][j] = sum_k B[k][j] = p_j + p_{j+16} (exact f32), held by lanes j and j+16.
// Remaining 16 -> 1 with 4 ds_swizzle xor steps. EXEC all-1s (uniform code path).
__device__ __forceinline__ float wave_reduce(float partial) {
    v2f a; a[0] = 1.0f;    a[1] = 1.0f;
    v2f b; b[0] = partial; b[1] = 0.0f;
    v8f c = {};
    v8f d = __builtin_amdgcn_wmma_f32_16x16x4_f32(
        /*neg_a=*/false, a, /*neg_b=*/false, b,
        /*c_mod=*/(short)0, c, /*reuse_a=*/false, /*reuse_b=*/false);
    float v = d[0];
    v = swz_add_1(v);
    v = swz_add_2(v);
    v = swz_add_4(v);
    v = swz_add_8(v);
    return v;   // full-wave sum, valid in every lane
}

__global__ __launch_bounds__(256) void pose_main(
    const float* __restrict__ pose3d,    // [3,21]
    const float* __restrict__ bone2d,    // [N,2,21]
    const float* __restrict__ Rd_all,    // [N,3,3]
    const float* __restrict__ Cd_all,    // [N,3,1]
    const float* __restrict__ Rcam,      // [3,3]
    float* __restrict__ block_out, int N)
{
    __shared__ float sp[63];      // pose3d staged (uniform, DS-broadcast reads)
    __shared__ float src[9];      // R_cam staged
    __shared__ float swave[8];

    const int tid = threadIdx.x;
    if (tid < 63)                 sp[tid]       = pose3d[tid];
    if (tid >= 64 && tid < 73)    src[tid - 64] = Rcam[tid - 64];
    __syncthreads();

    float rc[9];
    #pragma unroll
    for (int i = 0; i < 9; ++i) rc[i] = src[i];

    float partial = 0.0f;
    const int gstride = gridDim.x * blockDim.x;
    for (int n = blockIdx.x * blockDim.x + tid; n < N; n += gstride) {
        // --- per-sample rotation / translation ---
        const float* rd = Rd_all + (size_t)n * 9;
        float r[9];
        #pragma unroll
        for (int i = 0; i < 9; ++i) r[i] = rd[i];

        const float* cd = Cd_all + (size_t)n * 3;
        const float c0 = cd[0], c1 = cd[1], c2 = cd[2];

        // M[i][l] = sum_k Rcam[k][i] * Rd[l][k]   (P_cam = M @ (pose3d - C))
        float M[3][3];
        #pragma unroll
        for (int i = 0; i < 3; ++i)
            #pragma unroll
            for (int l = 0; l < 3; ++l)
                M[i][l] = fmaf(rc[0*3+i], r[l*3+0],
                          fmaf(rc[1*3+i], r[l*3+1],
                               rc[2*3+i] * r[l*3+2]));

        const float t0 = -(M[0][0]*c0 + M[0][1]*c1 + M[0][2]*c2);
        const float t1 = -(M[1][0]*c0 + M[1][1]*c1 + M[1][2]*c2);
        const float t2 = -(M[2][0]*c0 + M[2][1]*c1 + M[2][2]*c2);

        // --- bone_2d: 42 contiguous floats, 8B aligned -> b64 loads ---
        const float2* bp = reinterpret_cast<const float2*>(bone2d + (size_t)n * 42);
        float buf[42];
        #pragma unroll
        for (int k = 0; k < 21; ++k) {
            float2 v = bp[k];
            buf[2*k]     = v.x;
            buf[2*k + 1] = v.y;
        }

        float acc = 0.0f;
        #pragma unroll
        for (int j = 0; j < NJ; ++j) {
            const float X = sp[j], Y = sp[21 + j], Z = sp[42 + j];
            const float p0 = fmaf(M[0][0], X, fmaf(M[0][1], Y, fmaf(M[0][2], Z, t0)));
            const float p1 = fmaf(M[1][0], X, fmaf(M[1][1], Y, fmaf(M[1][2], Z, t1)));
            const float p2 = fmaf(M[2][0], X, fmaf(M[2][1], Y, fmaf(M[2][2], Z, t2)));
            const float zinv = __builtin_amdgcn_rcpf(p2);   // v_rcp_f32 (~1 ulp)
            const float dx = FOCAL_C * p0 * zinv + PX_C - buf[j];
            const float dy = FOCAL_C * p1 * zinv + PY_C - buf[21 + j];
            acc = fmaf(dx, dx, fmaf(dy, dy, acc));
        }
        partial += acc;
    }

    // --- block reduction: WMMA wave stage, then deterministic LDS fold ---
    const float wtot = wave_reduce(partial);
    const int wave = tid >> 5, lane = tid & 31;
    if (lane == 0) swave[wave] = wtot;
    __syncthreads();
    if (tid == 0) {
        float s = 0.0f;
        #pragma unroll
        for (int w = 0; w < 8; ++w) s += swave[w];
        block_out[blockIdx.x] = s;
    }
}

__global__ __launch_bounds__(256) void pose_finalize(
    const float* __restrict__ pose3d,
    const float* __restrict__ block_out,
    int nblocks, float inv_scale,
    float* __restrict__ out)
{
    __shared__ float red[256];
    const int tid = threadIdx.x;
    float s = 0.0f;
    for (int i = tid; i < nblocks; i += 256) s += block_out[i];
    red[tid] = s;
    __syncthreads();
    #pragma unroll
    for (int off = 128; off > 0; off >>= 1) {
        if (tid < off) red[tid] += red[tid + off];
        __syncthreads();
    }
    if (tid == 0) {
        const int LB[6][2] = {{0,1},{1,2},{2,3},{0,7},{7,8},{8,9}};
        const int RB[6][2] = {{0,4},{4,5},{5,6},{0,10},{10,11},{11,12}};
        float sym = 0.0f;
        #pragma unroll
        for (int b = 0; b < 6; ++b) {
            float ll = 0.0f, lr = 0.0f;
            #pragma unroll
            for (int cc = 0; cc < 3; ++cc) {
                const float dl = pose3d[cc*21 + LB[b][0]] - pose3d[cc*21 + LB[b][1]];
                const float dr = pose3d[cc*21 + RB[b][0]] - pose3d[cc*21 + RB[b][1]];
                ll = fmaf(dl, dl, ll);
                lr = fmaf(dr, dr, lr);
            }
            const float dd = ll - lr;
            sym = fmaf(dd, dd, sym);
        }
        sym *= (1.0f / 6.0f);
        out[0] = W_SYM * sym + W_PROJ * red[0] * inv_scale;
    }
}

extern "C" void kernel_launch(void* const* d_in, const int* in_sizes, int n_in,
                              void* d_out, int out_size, void* d_ws, size_t ws_size,
                              hipStream_t stream) {
    const float* pose3d = (const float*)d_in[0];
    const float* bone2d = (const float*)d_in[1];
    const float* Rd     = (const float*)d_in[2];
    const float* Cd     = (const float*)d_in[3];
    const float* Rc     = (const float*)d_in[4];

    const int N = in_sizes[1] / (2 * NJ);   // 262144
    const int threads = 256;
    int blocks = 1024;                       // 1 sample/thread at N=262144
    if ((long)blocks * threads > (long)N + threads - 1)
        blocks = (N + threads - 1) / threads;
    if (ws_size > 0 && (size_t)blocks * sizeof(float) > ws_size)
        blocks = (int)(ws_size / sizeof(float));
    if (blocks < 1) blocks = 1;

    float* block_out = (float*)d_ws;
    pose_main<<<blocks, threads, 0, stream>>>(pose3d, bone2d, Rd, Cd, Rc, block_out, N);

    const float inv_scale = 1.0f / ((float)N * 2.0f * (float)NJ);
    pose_finalize<<<1, 256, 0, stream>>>(pose3d, block_out, blocks, inv_scale, (float*)d_out);
}